// GAE_39599598469276
// MI455X (gfx1250) — compile-verified
//
#include <hip/hip_runtime.h>

typedef float v2f __attribute__((ext_vector_type(2)));
typedef float v8f __attribute__((ext_vector_type(8)));

#define GN 10000
#define GE 320000

// ---------------------------------------------------------------- utilities
__global__ void fill_kernel(float* p, long long n, float v) {
    long long i = (long long)blockIdx.x * blockDim.x + threadIdx.x;
    if (i < n) p[i] = v;
}

__global__ void degree_kernel(const int* __restrict__ src, const int* __restrict__ dst,
                              int E, float* degOut, float* degIn) {
    int i = blockIdx.x * blockDim.x + threadIdx.x;
    if (i >= E) return;
    atomicAdd(&degOut[src[i]], 1.0f);
    atomicAdd(&degIn[dst[i]], 1.0f);
}

__global__ void norm_kernel(float* d, int n) {
    int i = blockIdx.x * blockDim.x + threadIdx.x;
    if (i >= n) return;
    float v = d[i];
    d[i] = (v > 0.0f) ? rsqrtf(v) : 0.0f;
}

// -------------------------------------------------- WMMA fp32 GEMM: C = A@B
// Strip-mined: one wave computes NT adjacent 16x16 output tiles (a 16 x 16*NT
// strip).  The A fragment (2 VGPRs) is loaded once per k-step and reused for
// NT v_wmma_f32_16x16x4_f32 issues -> 4x less A traffic from L2.
template <int NT>
__global__ void gemm_wmma(const float* __restrict__ A, const float* __restrict__ B,
                          float* __restrict__ C, int M, int K, int Nc,
                          int accumulate, int relu) {
    int wid  = (blockIdx.x * blockDim.x + threadIdx.x) >> 5;
    int lane = threadIdx.x & 31;
    int tilesN = Nc >> 4;
    int strips = tilesN / NT;
    int tilesM = M >> 4;
    if (wid >= tilesM * strips) return;            // uniform per wave: EXEC all-1s
    int m0 = (wid / strips) << 4;
    int n0 = (wid % strips) * (NT << 4);
    int half = lane >> 4;                           // lanes 0-15: K=0,1 ; 16-31: K=2,3
    int lr   = lane & 15;
    const float* arow = A + (size_t)(m0 + lr) * K;
    v8f acc[NT];
#pragma unroll
    for (int t = 0; t < NT; ++t) acc[t] = (v8f){};
    for (int k0 = 0; k0 < K; k0 += 4) {
        int ka = k0 + (half << 1);
        v2f a;
        a.x = arow[ka];
        a.y = arow[ka + 1];
        const float* b0 = B + (size_t)ka * Nc + n0 + lr;
        const float* b1 = B + (size_t)(ka + 1) * Nc + n0 + lr;
#pragma unroll
        for (int t = 0; t < NT; ++t) {
            v2f b;
            b.x = b0[t * 16];
            b.y = b1[t * 16];
            acc[t] = __builtin_amdgcn_wmma_f32_16x16x4_f32(false, a, false, b,
                                                           (short)0, acc[t], false, false);
        }
    }
#pragma unroll
    for (int t = 0; t < NT; ++t) {
        int col = n0 + t * 16 + lr;
#pragma unroll
        for (int r = 0; r < 8; ++r) {
            int m = m0 + r + (half << 3);
            float v = acc[t][r];
            if (accumulate) v += C[(size_t)m * Nc + col];
            if (relu) v = fmaxf(v, 0.0f);
            C[(size_t)m * Nc + col] = v;
        }
    }
}

// ------------------------------------------------- edge scatter (atomics)
// agg[dst] += h[src] * normS[src], float4 per thread.  F multiple of 4.
__global__ void scatter_edges(const int* __restrict__ src, const int* __restrict__ dst,
                              const float* __restrict__ h, const float* __restrict__ normS,
                              float* agg, int E, int F4, int F) {
    long long idx = (long long)blockIdx.x * blockDim.x + threadIdx.x;
    long long tot = (long long)E * F4;
    if (idx >= tot) return;
    int e = (int)(idx / F4);
    int f = (int)(idx % F4) << 2;
    int s = src[e], d = dst[e];
    float ns = normS[s];
    const float4 v = *(const float4*)(h + (size_t)s * F + f);
    float* o = agg + (size_t)d * F + f;
    atomicAdd(o + 0, v.x * ns);
    atomicAdd(o + 1, v.y * ns);
    atomicAdd(o + 2, v.z * ns);
    atomicAdd(o + 3, v.w * ns);
}

// out = act((agg [+ selfH*normS]) * normD)
__global__ void finalize_kernel(const float* __restrict__ agg, const float* __restrict__ selfH,
                                const float* __restrict__ normS, const float* __restrict__ normD,
                                float* out, int n, int F, int relu) {
    int idx = blockIdx.x * blockDim.x + threadIdx.x;
    if (idx >= n) return;
    int i = idx / F;
    float v = agg[idx];
    if (selfH) v += selfH[idx] * normS[i];
    v *= normD[i];
    if (relu) v = fmaxf(v, 0.0f);
    out[idx] = v;
}

// row softmax over F=64, one wave32 per row (2 elems/lane)
__global__ void softmax_rows(float* X, int N) {
    int wave = threadIdx.x >> 5, lane = threadIdx.x & 31;
    int row = blockIdx.x * (blockDim.x >> 5) + wave;
    if (row >= N) return;
    float* x = X + (size_t)row * 64;
    float v0 = x[lane], v1 = x[lane + 32];
    float m = fmaxf(v0, v1);
    for (int o = 16; o; o >>= 1) m = fmaxf(m, __shfl_xor(m, o, 32));
    float e0 = __expf(v0 - m), e1 = __expf(v1 - m);
    float s = e0 + e1;
    for (int o = 16; o; o >>= 1) s += __shfl_xor(s, o, 32);
    float inv = 1.0f / s;
    x[lane] = e0 * inv;          // softmax >= 0, relu is identity
    x[lane + 32] = e1 * inv;
}

// zero-pad Wm_1 (32x10) -> (32x16)
__global__ void padw_kernel(const float* __restrict__ W, float* Wp) {
    int idx = blockIdx.x * blockDim.x + threadIdx.x;
    if (idx >= 32 * 16) return;
    int k = idx / 16, n = idx % 16;
    Wp[idx] = (n < 10) ? W[k * 10 + n] : 0.0f;
}

// xhPad (Nx16) -> out (Nx10)
__global__ void copy_xh(const float* __restrict__ xhPad, float* out, int N) {
    int idx = blockIdx.x * blockDim.x + threadIdx.x;
    if (idx >= N * 10) return;
    int i = idx / 10, f = idx % 10;
    out[idx] = xhPad[i * 16 + f];
}

// ---------------------------------------- decoder: adj = xh @ xh^T  (K=16)
// One wave per 16x16 tile; nontemporal stores keep the 400MB adj out of L2
// (everything else in the pipeline is L2-resident and benefits).
__global__ void decoder_wmma(const float* __restrict__ xh, float* __restrict__ adj, int N) {
    int wave = threadIdx.x >> 5, lane = threadIdx.x & 31;
    int nt = blockIdx.x * 8 + wave;
    int mt = blockIdx.y;
    int tiles = N >> 4;                     // 625
    if (nt >= tiles) return;                // uniform per wave
    int half = lane >> 4, lr = lane & 15;
    const float* arow = xh + (size_t)(mt * 16 + lr) * 16;
    const float* brow = xh + (size_t)(nt * 16 + lr) * 16;
    v8f acc = {};
#pragma unroll
    for (int k0 = 0; k0 < 16; k0 += 4) {
        int ka = k0 + (half << 1);
        v2f a, b;
        a.x = arow[ka]; a.y = arow[ka + 1];
        b.x = brow[ka]; b.y = brow[ka + 1];
        acc = __builtin_amdgcn_wmma_f32_16x16x4_f32(false, a, false, b,
                                                    (short)0, acc, false, false);
    }
    int col = nt * 16 + lr;
#pragma unroll
    for (int r = 0; r < 8; ++r) {
        int m = mt * 16 + r + (half << 3);
        __builtin_nontemporal_store(acc[r], &adj[(size_t)m * N + col]);
    }
}

// ---------------------------------------------------------------- launcher
extern "C" void kernel_launch(void* const* d_in, const int* in_sizes, int n_in,
                              void* d_out, int out_size, void* d_ws, size_t ws_size,
                              hipStream_t stream) {
    const int N = GN, E = GE;
    const float* feat[3] = { (const float*)d_in[0], (const float*)d_in[1], (const float*)d_in[2] };
    const int*   edg[4]  = { (const int*)d_in[3], (const int*)d_in[4],
                             (const int*)d_in[5], (const int*)d_in[6] };
    const float* Wa[3] = { (const float*)d_in[7],  (const float*)d_in[9],  (const float*)d_in[11] };
    const float* Wb[3] = { (const float*)d_in[8],  (const float*)d_in[10], (const float*)d_in[12] };
    const float* Wf[3] = { (const float*)d_in[13], (const float*)d_in[14], (const float*)d_in[15] };
    const float* Wm0 = (const float*)d_in[16];
    const float* Wm1 = (const float*)d_in[17];
    float* out = (float*)d_out;

    // workspace layout (floats)
    float* ws    = (float*)d_ws;
    float* normS = ws;                         // N
    float* normD = normS + N;                  // N
    float* bufH  = normD + N;                  // N*256
    float* bufA  = bufH + (size_t)N * 256;     // N*256
    float* z0    = bufA + (size_t)N * 256;     // N*64
    float* z1    = z0 + (size_t)N * 64;
    float* z2    = z1 + (size_t)N * 64;
    float* fus   = z2 + (size_t)N * 64;        // N*64
    float* xhPad = fus + (size_t)N * 64;       // N*16
    float* Wpad  = xhPad + (size_t)N * 16;     // 512
    float* zv[3] = { z0, z1, z2 };

    auto cdiv = [](long long a, long long b) { return (int)((a + b - 1) / b); };
    auto gemm = [&](const float* A, const float* B, float* C, int M, int K, int Nc,
                    int acc, int relu) {
        int tilesM = M >> 4, tilesN = Nc >> 4;
        if ((tilesN & 3) == 0) {
            int waves = tilesM * (tilesN >> 2);
            gemm_wmma<4><<<cdiv(waves, 8), 256, 0, stream>>>(A, B, C, M, K, Nc, acc, relu);
        } else if ((tilesN & 1) == 0) {
            int waves = tilesM * (tilesN >> 1);
            gemm_wmma<2><<<cdiv(waves, 8), 256, 0, stream>>>(A, B, C, M, K, Nc, acc, relu);
        } else {
            int waves = tilesM * tilesN;
            gemm_wmma<1><<<cdiv(waves, 8), 256, 0, stream>>>(A, B, C, M, K, Nc, acc, relu);
        }
    };

    // ---- per-view 2-layer GCN stacks ----
    for (int v = 0; v < 3; ++v) {
        const int* src = edg[v];
        const int* dst = edg[v] + E;
        fill_kernel<<<cdiv(2 * N, 256), 256, 0, stream>>>(normS, 2 * N, 0.0f);
        degree_kernel<<<cdiv(E, 256), 256, 0, stream>>>(src, dst, E, normS, normD);
        norm_kernel<<<cdiv(2 * N, 256), 256, 0, stream>>>(normS, 2 * N);
        // layer 1: 512 -> 256, relu
        gemm(feat[v], Wa[v], bufH, N, 512, 256, 0, 0);
        fill_kernel<<<cdiv((long long)N * 256, 256), 256, 0, stream>>>(bufA, (long long)N * 256, 0.0f);
        scatter_edges<<<cdiv((long long)E * 64, 256), 256, 0, stream>>>(src, dst, bufH, normS, bufA, E, 64, 256);
        finalize_kernel<<<cdiv((long long)N * 256, 256), 256, 0, stream>>>(bufA, nullptr, normS, normD, bufH, N * 256, 256, 1);
        // layer 2: 256 -> 64, identity
        gemm(bufH, Wb[v], bufA, N, 256, 64, 0, 0);
        fill_kernel<<<cdiv((long long)N * 64, 256), 256, 0, stream>>>(zv[v], (long long)N * 64, 0.0f);
        scatter_edges<<<cdiv((long long)E * 16, 256), 256, 0, stream>>>(src, dst, bufA, normS, zv[v], E, 16, 64);
        finalize_kernel<<<cdiv((long long)N * 64, 256), 256, 0, stream>>>(zv[v], nullptr, normS, normD, zv[v], N * 64, 64, 0);
    }

    // ---- fusion: softmax(z0@Wf0 + z1@Wf1 + z2@Wf2) then relu ----
    gemm(z0, Wf[0], fus, N, 64, 64, 0, 0);
    gemm(z1, Wf[1], fus, N, 64, 64, 1, 0);
    gemm(z2, Wf[2], fus, N, 64, 64, 1, 0);
    softmax_rows<<<cdiv(N, 8), 256, 0, stream>>>(fus, N);

    // ---- shared stack on graph-with-self-loops ----
    {
        const int* src = edg[3];
        const int* dst = edg[3] + E;
        fill_kernel<<<cdiv(2 * N, 256), 256, 0, stream>>>(normS, 2 * N, 1.0f);  // self loop
        degree_kernel<<<cdiv(E, 256), 256, 0, stream>>>(src, dst, E, normS, normD);
        norm_kernel<<<cdiv(2 * N, 256), 256, 0, stream>>>(normS, 2 * N);
        // layer: 64 -> 32, relu (self-loop folded into finalize)
        gemm(fus, Wm0, bufH, N, 64, 32, 0, 0);
        fill_kernel<<<cdiv((long long)N * 32, 256), 256, 0, stream>>>(bufA, (long long)N * 32, 0.0f);
        scatter_edges<<<cdiv((long long)E * 8, 256), 256, 0, stream>>>(src, dst, bufH, normS, bufA, E, 8, 32);
        finalize_kernel<<<cdiv((long long)N * 32, 256), 256, 0, stream>>>(bufA, bufH, normS, normD, fus, N * 32, 32, 1);
        // layer: 32 -> 10 (padded to 16), identity
        padw_kernel<<<2, 256, 0, stream>>>(Wm1, Wpad);
        gemm(fus, Wpad, bufH, N, 32, 16, 0, 0);
        fill_kernel<<<cdiv((long long)N * 16, 256), 256, 0, stream>>>(xhPad, (long long)N * 16, 0.0f);
        scatter_edges<<<cdiv((long long)E * 4, 256), 256, 0, stream>>>(src, dst, bufH, normS, xhPad, E, 4, 16);
        finalize_kernel<<<cdiv((long long)N * 16, 256), 256, 0, stream>>>(xhPad, bufH, normS, normD, xhPad, N * 16, 16, 0);
    }

    // ---- outputs: xh then adj = xh @ xh^T ----
    copy_xh<<<cdiv((long long)N * 10, 256), 256, 0, stream>>>(xhPad, out + (size_t)N * N, N);
    decoder_wmma<<<dim3(cdiv(N >> 4, 8), N >> 4), 256, 0, stream>>>(xhPad, out, N);
}